// GATEncoder_44401371906307
// MI455X (gfx1250) — compile-verified
//
#include <hip/hip_runtime.h>

// ---------------------------------------------------------------------------
// GAT encoder for MI455X (gfx1250). Dense GEMMs use V_WMMA_F32_16X16X4_F32
// (fp32 matrix pipe, full precision). Edge softmax/aggregation use atomics.
// ---------------------------------------------------------------------------

#define NN    20000
#define EE    320000
#define DIN   64
#define HIDC  32
#define HEADS 8
#define OUTC  128
#define DMID  256            // HEADS*HIDC
#define D2    1024           // HEADS*OUTC
#define NEDGE (EE + NN)      // edges + self loops

typedef float v2f  __attribute__((ext_vector_type(2)));
typedef float v8f_t __attribute__((ext_vector_type(8)));

static inline int cdiv(int a, int b) { return (a + b - 1) / b; }

// ---- order-preserving float <-> uint encoding for atomic max ---------------
__device__ __forceinline__ unsigned f2o(float f) {
    unsigned u = __float_as_uint(f);
    return (u & 0x80000000u) ? ~u : (u | 0x80000000u);
}
__device__ __forceinline__ float o2f(unsigned e) {
    unsigned u = (e & 0x80000000u) ? (e ^ 0x80000000u) : ~e;
    return __uint_as_float(u);
}

__device__ __forceinline__ void edge_sd(const int* ei, int e, int& s, int& d) {
    if (e < EE) { s = ei[e]; d = ei[EE + e]; }
    else        { s = e - EE; d = e - EE; }      // self loop
}

__device__ __forceinline__ float lrelu02(float v) { return v > 0.f ? v : 0.2f * v; }
__device__ __forceinline__ float eluf(float v)    { return v > 0.f ? v : (expf(v) - 1.f); }
__device__ __forceinline__ float sigm(float v)    { return 1.f / (1.f + expf(-v)); }

// ---------------------------------------------------------------------------
// fills
// ---------------------------------------------------------------------------
__global__ void fill_f32(float* p, float v, int n) {
    int i = blockIdx.x * blockDim.x + threadIdx.x;
    if (i < n) p[i] = v;
}
__global__ void fill_u32(unsigned* p, unsigned v, int n) {
    int i = blockIdx.x * blockDim.x + threadIdx.x;
    if (i < n) p[i] = v;
}

// ---------------------------------------------------------------------------
// WMMA fp32 GEMM: O[N x M] = A[N x K] @ B[K x M]. One wave -> one 16x16 tile.
// grid = (M/16, N/16), block = 32 (one wave32, EXEC all ones for WMMA).
// A-matrix lane layout (16x4 f32): lane m = L&15, khalf = L>>4,
//   v[0]=A[m][k+2*khalf], v[1]=A[m][k+2*khalf+1].
// B-matrix (4x16): v[0]=B[k+2*khalf][n], v[1]=B[k+2*khalf+1][n], n = L&15.
// C/D (16x16): acc[i] -> row (8*khalf + i), col n.
// ---------------------------------------------------------------------------
__global__ void wmma_gemm_f32(const float* __restrict__ A,
                              const float* __restrict__ B,
                              float* __restrict__ O, int K, int M) {
    const int lane = threadIdx.x;
    const int m  = lane & 15;
    const int kh = lane >> 4;
    const long long row0 = (long long)blockIdx.y * 16;
    const int col0 = blockIdx.x * 16;

    const float* __restrict__ arow = A + (row0 + m) * K;
    v8f_t acc = {};
    for (int k = 0; k < K; k += 4) {
        v2f a, b;
        a.x = arow[k + 2 * kh];
        a.y = arow[k + 2 * kh + 1];
        b.x = B[(long long)(k + 2 * kh)     * M + col0 + m];
        b.y = B[(long long)(k + 2 * kh + 1) * M + col0 + m];
        acc = __builtin_amdgcn_wmma_f32_16x16x4_f32(
            /*neg_a=*/false, a, /*neg_b=*/false, b,
            /*c_mod=*/(short)0, acc, /*reuse_a=*/false, /*reuse_b=*/false);
    }
#pragma unroll
    for (int i = 0; i < 8; ++i)
        O[(row0 + kh * 8 + i) * M + col0 + m] = acc[i];
}

// ---------------------------------------------------------------------------
// attention logits: als[n,h] = <hf[n,h,:], a_s[h,:]>, likewise ald.
// ---------------------------------------------------------------------------
__global__ void att_logits_k(const float* __restrict__ hf,
                             const float* __restrict__ a_s,
                             const float* __restrict__ a_d,
                             float* __restrict__ als, float* __restrict__ ald,
                             int C) {
    int t = blockIdx.x * blockDim.x + threadIdx.x;
    if (t >= NN * HEADS) return;
    int h = t & (HEADS - 1);
    int n = t >> 3;
    const float* hp = hf + ((long long)n * HEADS + h) * C;
    const float* ap = a_s + h * C;
    const float* bp = a_d + h * C;
    float s = 0.f, d = 0.f;
    for (int c = 0; c < C; ++c) { float v = hp[c]; s += v * ap[c]; d += v * bp[c]; }
    als[t] = s; ald[t] = d;
}

// ---------------------------------------------------------------------------
// edge pass 1: segment max of leaky_relu(als[src]+ald[dst]) into menc (encoded)
// ---------------------------------------------------------------------------
__global__ void edge_max_k(const int* __restrict__ ei,
                           const float* __restrict__ als,
                           const float* __restrict__ ald,
                           unsigned* __restrict__ menc) {
    int t = blockIdx.x * blockDim.x + threadIdx.x;
    if (t >= NEDGE * HEADS) return;
    int h = t & (HEADS - 1);
    int e = t >> 3;
    int s, d; edge_sd(ei, e, s, d);
    float v = lrelu02(als[s * HEADS + h] + ald[d * HEADS + h]);
    atomicMax(&menc[d * HEADS + h], f2o(v));
}

// ---------------------------------------------------------------------------
// edge pass 2: ex = exp(e - max[dst]); denom[dst] += ex
// ---------------------------------------------------------------------------
__global__ void edge_exp_k(const int* __restrict__ ei,
                           const float* __restrict__ als,
                           const float* __restrict__ ald,
                           const unsigned* __restrict__ menc,
                           float* __restrict__ den, float* __restrict__ ex) {
    int t = blockIdx.x * blockDim.x + threadIdx.x;
    if (t >= NEDGE * HEADS) return;
    int h = t & (HEADS - 1);
    int e = t >> 3;
    int s, d; edge_sd(ei, e, s, d);
    float v = lrelu02(als[s * HEADS + h] + ald[d * HEADS + h]);
    float m = o2f(menc[d * HEADS + h]);
    float x = expf(v - m);
    ex[t] = x;
    atomicAdd(&den[d * HEADS + h], x);
}

// ---------------------------------------------------------------------------
// edge pass 3: agg[dst,h,:] += (ex/denom) * hf[src,h,:]  (channel-quad threads)
// ---------------------------------------------------------------------------
__global__ void edge_agg_k(const int* __restrict__ ei,
                           const float* __restrict__ ex,
                           const float* __restrict__ den,
                           const float* __restrict__ hf,
                           float* __restrict__ agg, int C4, int total) {
    int t = blockIdx.x * blockDim.x + threadIdx.x;
    if (t >= total) return;
    int c4 = t % C4;
    int eh = t / C4;
    int h = eh & (HEADS - 1);
    int e = eh >> 3;
    int s, d; edge_sd(ei, e, s, d);
    float alpha = ex[eh] / (den[d * HEADS + h] + 1e-16f);
    const int C = C4 * 4;
    const float4 hv = *(const float4*)(hf + ((long long)s * HEADS + h) * C + c4 * 4);
    float* ap = agg + ((long long)d * HEADS + h) * C + c4 * 4;
    atomicAdd(ap + 0, alpha * hv.x);
    atomicAdd(ap + 1, alpha * hv.y);
    atomicAdd(ap + 2, alpha * hv.z);
    atomicAdd(ap + 3, alpha * hv.w);
}

// ---------------------------------------------------------------------------
// epilogue for concat layers (D=256): h = elu(LN(agg+b) + sig(rw)*residual)
// mode 0: residual = res[n,c] + rb[c]   (layer 0 projection)
// mode 1: residual = hprev[n,c]          (layer 1 identity)
// one block (256 threads) per node.
// ---------------------------------------------------------------------------
__global__ void epilogue01_k(const float* __restrict__ agg,
                             const float* __restrict__ bias,
                             const float* __restrict__ lng,
                             const float* __restrict__ lnb,
                             const float* __restrict__ res,
                             const float* __restrict__ rb,
                             const float* __restrict__ rw,
                             const float* __restrict__ hprev,
                             float* __restrict__ hout, int mode) {
    int n = blockIdx.x, c = threadIdx.x;
    float g = agg[n * DMID + c] + bias[c];
    __shared__ float s1[DMID], s2[DMID];
    s1[c] = g; s2[c] = g * g;
    __syncthreads();
    for (int s = DMID / 2; s > 0; s >>= 1) {
        if (c < s) { s1[c] += s1[c + s]; s2[c] += s2[c + s]; }
        __syncthreads();
    }
    float mu  = s1[0] * (1.f / DMID);
    float var = s2[0] * (1.f / DMID) - mu * mu;
    float ln  = (g - mu) * rsqrtf(var + 1e-5f) * lng[c] + lnb[c];
    float sg  = sigm(rw[0]);
    float r   = (mode == 0) ? (res[n * DMID + c] + rb[c]) : hprev[n * DMID + c];
    hout[n * DMID + c] = eluf(ln + sg * r);
}

// ---------------------------------------------------------------------------
// final epilogue (D=128): out = LN(mean_h(agg)+b2) + sig(rw2)*(res+rb2)
// one block (128 threads) per node.
// ---------------------------------------------------------------------------
__global__ void epilogue2_k(const float* __restrict__ agg,
                            const float* __restrict__ b2,
                            const float* __restrict__ lng,
                            const float* __restrict__ lnb,
                            const float* __restrict__ res,
                            const float* __restrict__ rb,
                            const float* __restrict__ rw,
                            float* __restrict__ out) {
    int n = blockIdx.x, c = threadIdx.x;
    float s = 0.f;
#pragma unroll
    for (int h = 0; h < HEADS; ++h)
        s += agg[((long long)n * HEADS + h) * OUTC + c];
    float g = s * (1.f / HEADS) + b2[c];
    __shared__ float s1[OUTC], s2[OUTC];
    s1[c] = g; s2[c] = g * g;
    __syncthreads();
    for (int t = OUTC / 2; t > 0; t >>= 1) {
        if (c < t) { s1[c] += s1[c + t]; s2[c] += s2[c + t]; }
        __syncthreads();
    }
    float mu  = s1[0] * (1.f / OUTC);
    float var = s2[0] * (1.f / OUTC) - mu * mu;
    float ln  = (g - mu) * rsqrtf(var + 1e-5f) * lng[c] + lnb[c];
    out[n * OUTC + c] = ln + sigm(rw[0]) * (res[n * OUTC + c] + rb[c]);
}

// ---------------------------------------------------------------------------
// orchestration
// ---------------------------------------------------------------------------
extern "C" void kernel_launch(void* const* d_in, const int* in_sizes, int n_in,
                              void* d_out, int out_size, void* d_ws, size_t ws_size,
                              hipStream_t stream) {
    (void)in_sizes; (void)n_in; (void)out_size; (void)ws_size;

    const float* x    = (const float*)d_in[0];
    const int*   ei   = (const int*)  d_in[1];
    const float* W0   = (const float*)d_in[2];
    const float* b0   = (const float*)d_in[3];
    const float* as0  = (const float*)d_in[4];
    const float* ad0  = (const float*)d_in[5];
    const float* lng0 = (const float*)d_in[6];
    const float* lnb0 = (const float*)d_in[7];
    const float* rW0  = (const float*)d_in[8];
    const float* rb0  = (const float*)d_in[9];
    const float* rw0  = (const float*)d_in[10];
    const float* W1   = (const float*)d_in[11];
    const float* b1   = (const float*)d_in[12];
    const float* as1  = (const float*)d_in[13];
    const float* ad1  = (const float*)d_in[14];
    const float* lng1 = (const float*)d_in[15];
    const float* lnb1 = (const float*)d_in[16];
    const float* rw1  = (const float*)d_in[17];
    const float* W2   = (const float*)d_in[18];
    const float* b2   = (const float*)d_in[19];
    const float* as2  = (const float*)d_in[20];
    const float* ad2  = (const float*)d_in[21];
    const float* lng2 = (const float*)d_in[22];
    const float* lnb2 = (const float*)d_in[23];
    const float* rW2  = (const float*)d_in[24];
    const float* rb2  = (const float*)d_in[25];
    const float* rw2  = (const float*)d_in[26];
    float* out = (float*)d_out;

    // workspace carve-up (256B aligned)
    char* ws = (char*)d_ws;
    size_t off = 0;
    auto carve = [&](size_t bytes) {
        void* p = ws + off;
        off += (bytes + 255) & ~(size_t)255;
        return p;
    };
    float*    HF   = (float*)   carve((size_t)NN * D2 * 4);      // transformed feats (<= N x 1024)
    float*    AGG  = (float*)   carve((size_t)NN * D2 * 4);      // aggregation
    float*    H    = (float*)   carve((size_t)NN * DMID * 4);    // layer activations
    float*    RES  = (float*)   carve((size_t)NN * DMID * 4);    // residual projections
    float*    ALS  = (float*)   carve((size_t)NN * HEADS * 4);
    float*    ALD  = (float*)   carve((size_t)NN * HEADS * 4);
    unsigned* MAXE = (unsigned*)carve((size_t)NN * HEADS * 4);
    float*    DEN  = (float*)   carve((size_t)NN * HEADS * 4);
    float*    EX   = (float*)   carve((size_t)NEDGE * HEADS * 4);

    const int nh   = NN * HEADS;            // 160k
    const int neh  = NEDGE * HEADS;         // 2.72M
    const dim3 blk(256);

    auto run_edges = [&](const float* a_s, const float* a_d, int C, int aggElems) {
        att_logits_k<<<cdiv(nh, 256), blk, 0, stream>>>(HF, a_s, a_d, ALS, ALD, C);
        fill_u32<<<cdiv(nh, 256), blk, 0, stream>>>(MAXE, 0u, nh);
        fill_f32<<<cdiv(nh, 256), blk, 0, stream>>>(DEN, 0.f, nh);
        fill_f32<<<cdiv(aggElems, 256), blk, 0, stream>>>(AGG, 0.f, aggElems);
        edge_max_k<<<cdiv(neh, 256), blk, 0, stream>>>(ei, ALS, ALD, MAXE);
        edge_exp_k<<<cdiv(neh, 256), blk, 0, stream>>>(ei, ALS, ALD, MAXE, DEN, EX);
        const int C4 = C / 4;
        const int total = neh * C4;
        edge_agg_k<<<cdiv(total, 256), blk, 0, stream>>>(ei, EX, DEN, HF, AGG, C4, total);
    };

    // ---------------- layer 0: GAT(64 -> 8x32 concat) ----------------
    wmma_gemm_f32<<<dim3(DMID / 16, NN / 16), dim3(32), 0, stream>>>(x,  W0,  HF,  DIN, DMID);
    wmma_gemm_f32<<<dim3(DMID / 16, NN / 16), dim3(32), 0, stream>>>(x,  rW0, RES, DIN, DMID);
    run_edges(as0, ad0, HIDC, NN * DMID);
    epilogue01_k<<<NN, DMID, 0, stream>>>(AGG, b0, lng0, lnb0, RES, rb0, rw0, nullptr, H, 0);

    // ---------------- layer 1: GAT(256 -> 8x32 concat), identity residual ----
    wmma_gemm_f32<<<dim3(DMID / 16, NN / 16), dim3(32), 0, stream>>>(H, W1, HF, DMID, DMID);
    run_edges(as1, ad1, HIDC, NN * DMID);
    epilogue01_k<<<NN, DMID, 0, stream>>>(AGG, b1, lng1, lnb1, nullptr, nullptr, rw1, H, H, 1);

    // ---------------- layer 2: GAT(256 -> 8x128 mean) ----------------
    wmma_gemm_f32<<<dim3(D2 / 16,   NN / 16), dim3(32), 0, stream>>>(H, W2,  HF,  DMID, D2);
    wmma_gemm_f32<<<dim3(OUTC / 16, NN / 16), dim3(32), 0, stream>>>(H, rW2, RES, DMID, OUTC);
    run_edges(as2, ad2, OUTC, NN * D2);
    epilogue2_k<<<NN, OUTC, 0, stream>>>(AGG, b2, lng2, lnb2, RES, rb2, rw2, out);
}